// PatchEmbedder_59227599012059
// MI455X (gfx1250) — compile-verified
//
#include <hip/hip_runtime.h>
#include <hip/hip_bf16.h>

// ---------------------------------------------------------------------------
// Complex PatchEmbedder for MI455X (gfx1250, wave32, WMMA).
//   LN1(whiten) -> patch conv (complex GEMM, v_wmma_f32_16x16x4_f32) -> LN2 -> transpose
// CDNA5 paths: v_wmma_f32_16x16x4_f32, global_load_async_to_lds_b128 (+ASYNCcnt),
//              global_prefetch_b8.
// ---------------------------------------------------------------------------

typedef float v2f __attribute__((ext_vector_type(2)));
typedef float v8f __attribute__((ext_vector_type(8)));

namespace pe {
constexpr int Cc   = 3;
constexpr int Hh   = 384;
constexpr int Ww   = 384;
constexpr int Pp   = 16;
constexpr int Ee   = 768;
constexpr int HP   = 24;        // H / P
constexpr int WP   = 24;
constexpr int MPB  = HP * WP;   // 576 patches per batch
constexpr int Bb   = 64;
constexpr int Mm   = Bb * MPB;  // 36864
constexpr int Kk   = Cc * Pp * Pp;  // 768 (complex K)
constexpr int CHW  = Cc * Hh * Ww;  // 442368
constexpr int EHW  = Ee * MPB;      // 442368 (same count as CHW!)
constexpr int NSPLIT = 16;
constexpr int CHUNK  = CHW / NSPLIT; // 27648
constexpr float EPS = 1e-5f;
} // namespace pe

__device__ __forceinline__ v8f wmma_f32(v2f a, v2f b, v8f c) {
  // V_WMMA_F32_16X16X4_F32: f32 WMMA, A 16x4, B 4x16, C/D 16x16 f32.
  // f32 WMMA only supports CNeg in NEG -> neg_a/neg_b must be false.
  return __builtin_amdgcn_wmma_f32_16x16x4_f32(
      /*neg_a=*/false, a, /*neg_b=*/false, b,
      /*c_mod=*/(short)0, c, /*reuse_a=*/false, /*reuse_b=*/false);
}

// Derive whitening params (mean_r, mean_i, wrr, wii, wri) from raw moment sums.
__device__ __forceinline__ void derive_whiten(const float s[5], float inv_n,
                                              float& mr, float& mi,
                                              float& wrr, float& wii, float& wri) {
  mr = s[0] * inv_n;
  mi = s[1] * inv_n;
  float vrr = s[2] * inv_n - mr * mr + pe::EPS;
  float vii = s[3] * inv_n - mi * mi + pe::EPS;
  float vri = s[4] * inv_n - mr * mi;
  float det = vrr * vii - vri * vri;
  float sq  = sqrtf(det);
  float tt  = sqrtf(vrr + vii + 2.0f * sq);
  float ist = 1.0f / (sq * tt);
  wrr = (vii + sq) * ist;
  wii = (vrr + sq) * ist;
  wri = -vri * ist;
}

// ---------------------------------------------------------------------------
// K1/K3: raw moment partial sums over one batch-slab of complex data.
// grid = Bb * NSPLIT blocks, 256 threads. Writes partial[(b*NSPLIT+s)*5 + q].
// Deterministic (fixed-order tree reduction, no float atomics).
// ---------------------------------------------------------------------------
__global__ void pe_stats_kernel(const float2* __restrict__ src,
                                float* __restrict__ partial) {
  const int b = blockIdx.x / pe::NSPLIT;
  const int s = blockIdx.x % pe::NSPLIT;
  const int t = threadIdx.x;
  const size_t base = (size_t)b * pe::CHW + (size_t)s * pe::CHUNK;

  float acc[5] = {0.f, 0.f, 0.f, 0.f, 0.f};
  constexpr int NITER = pe::CHUNK / 256; // 108
#pragma unroll 4
  for (int j = 0; j < NITER; ++j) {
    if (j + 16 < NITER) {
      // gfx1250 speculative prefetch (global_prefetch_b8)
      __builtin_prefetch(&src[base + t + 256 * (j + 16)], 0, 1);
    }
    float2 v = src[base + t + 256 * j];
    acc[0] += v.x;
    acc[1] += v.y;
    acc[2] += v.x * v.x;
    acc[3] += v.y * v.y;
    acc[4] += v.x * v.y;
  }

  __shared__ float red[5][256];
#pragma unroll
  for (int q = 0; q < 5; ++q) red[q][t] = acc[q];
  __syncthreads();
  for (int off = 128; off > 0; off >>= 1) {
    if (t < off) {
#pragma unroll
      for (int q = 0; q < 5; ++q) red[q][t] += red[q][t + off];
    }
    __syncthreads();
  }
  if (t == 0) {
#pragma unroll
    for (int q = 0; q < 5; ++q)
      partial[(size_t)(b * pe::NSPLIT + s) * 5 + q] = red[q][0];
  }
}

// ---------------------------------------------------------------------------
// K2: fused LN1-apply + complex patch-GEMM + conv bias.
// Output written straight into d_out as [B, Hp*Wp, E] complex (final layout).
// WG = 256 threads (8 waves). Tile: 32 (M) x 64 (N). Each wave owns one
// 16x16 complex tile (two v8f accumulators). K staged in LDS 32 at a time.
// A tile: LN1 applied on the fly, stored as split re/im planes (8B-aligned
//         v2f rows, stride 34 floats -> conflict-free, mov-free fragments).
// B tile: raw conv_w bytes DMA'd with global_load_async_to_lds_b128
//         (ASYNCcnt), 16B-aligned rows (stride 34 float2 = 272B).
// ---------------------------------------------------------------------------
__global__ void pe_gemm_ln1_kernel(const float2* __restrict__ x,
                                   const float*  __restrict__ g1,   // [4][C,H,W]
                                   const float2* __restrict__ b1,   // [C,H,W]
                                   const float2* __restrict__ wgt,  // [E, K] complex
                                   const float2* __restrict__ cb,   // [E]
                                   const float*  __restrict__ part1,
                                   float2* __restrict__ out) {
  constexpr int NT = pe::Ee / 64;        // 12 N-tiles
  const int tn = blockIdx.x % NT;        // innermost -> A-tile reuse in L2
  const int tm = blockIdx.x / NT;
  const int m0 = tm * 32;
  const int n0 = tn * 64;
  const int b  = m0 / pe::MPB;           // 576 % 32 == 0 -> tile in one batch
  const int t  = threadIdx.x;

  __shared__ v2f    sAr[32][17];         // row stride 34 floats (8B aligned)
  __shared__ v2f    sAi[32][17];
  __shared__ float2 sB[64][34];          // row stride 272B (16B aligned)
  __shared__ float  sParams[5];          // mr, mi, wrr, wii, wri

  if (t == 0) {
    float s[5] = {0.f, 0.f, 0.f, 0.f, 0.f};
    for (int i = 0; i < pe::NSPLIT; ++i)
#pragma unroll
      for (int q = 0; q < 5; ++q)
        s[q] += part1[(size_t)(b * pe::NSPLIT + i) * 5 + q];
    float mr, mi, wrr, wii, wri;
    derive_whiten(s, 1.0f / (float)pe::CHW, mr, mi, wrr, wii, wri);
    sParams[0] = mr; sParams[1] = mi;
    sParams[2] = wrr; sParams[3] = wii; sParams[4] = wri;
  }
  __syncthreads();
  const float mr  = sParams[0], mi  = sParams[1];
  const float wrr = sParams[2], wii = sParams[3], wri = sParams[4];

  const int lane = t & 31;
  const int wv   = t >> 5;           // wave 0..7
  const int mw   = (wv >> 2) * 16;   // 0 or 16
  const int nw   = (wv & 3) * 16;    // 0,16,32,48
  const int ln   = lane & 15;
  const int kh   = lane >> 4;        // K sub-pair select: 0 -> {k,k+1}, 1 -> {k+2,k+3}

  v8f accR = {0.f, 0.f, 0.f, 0.f, 0.f, 0.f, 0.f, 0.f};
  v8f accI = {0.f, 0.f, 0.f, 0.f, 0.f, 0.f, 0.f, 0.f};

  const int k_l   = t & 31;          // this thread's K column in the A tile
  const int mrow0 = t >> 5;          // base row for A fill

  float* fAr = (float*)&sAr[0][0];
  float* fAi = (float*)&sAi[0][0];
  const unsigned sb_base = (unsigned)(uintptr_t)&sB[0][0];

  for (int kc = 0; kc < pe::Kk; kc += 32) {
    // ---- B tile: async DMA global -> LDS (raw bytes, ASYNCcnt tracked) ----
#pragma unroll
    for (int i = 0; i < 4; ++i) {
      const int idx = t + 256 * i;         // 1024 x b128
      const int kp  = idx & 15;            // 16B chunk within row
      const int n   = idx >> 4;            // 0..63
      const unsigned goff =
          (unsigned)((((n0 + n) * pe::Kk) + kc + (kp << 1)) * (int)sizeof(float2));
      const unsigned laddr =
          sb_base + (unsigned)((n * 34 + (kp << 1)) * (int)sizeof(float2));
      asm volatile("global_load_async_to_lds_b128 %0, %1, %2"
                   :: "v"(laddr), "v"(goff), "s"(wgt)
                   : "memory");
    }

    // ---- A tile: LN1-whitened, affine-transformed patch data ----
    // decode K coordinate once per kc (constant across i-loop)
    const int k   = kc + k_l;
    const int c   = k >> 8;          // K = 768 = C(3) * 256
    const int rem = k & 255;
    const int py  = rem >> 4;
    const int px  = rem & 15;

#pragma unroll
    for (int i = 0; i < 4; ++i) {
      const int m_l = mrow0 + 8 * i;
      const int m   = m0 + m_l;
      const int p   = m - b * pe::MPB;
      const int hp  = p / pe::WP;
      const int wp  = p % pe::WP;
      const int h   = hp * pe::Pp + py;
      const int w   = wp * pe::Pp + px;
      const size_t xoff = ((size_t)(b * pe::Cc + c) * pe::Hh + h) * pe::Ww + w;
      const size_t goffp = ((size_t)c * pe::Hh + h) * pe::Ww + w;
      float2 xv = x[xoff];
      float xr = xv.x - mr;
      float xi = xv.y - mi;
      float zr = wrr * xr + wri * xi;
      float zi = wri * xr + wii * xi;
      float g00 = g1[goffp];
      float g01 = g1[(size_t)pe::CHW + goffp];
      float g10 = g1[2 * (size_t)pe::CHW + goffp];
      float g11 = g1[3 * (size_t)pe::CHW + goffp];
      float2 bv = b1[goffp];
      fAr[m_l * 34 + k_l] = g00 * zr + g01 * zi + bv.x;
      fAi[m_l * 34 + k_l] = g10 * zr + g11 * zi + bv.y;
    }

    // DMA completion for this wave's async issues, then WG-wide visibility.
    asm volatile("s_wait_asynccnt 0" ::: "memory");
    __syncthreads();

    // ---- 8 WMMA K-steps of 4 ----
#pragma unroll
    for (int k4 = 0; k4 < 32; k4 += 4) {
      const int kk  = k4 + 2 * kh;             // even
      const int kk2 = (k4 >> 1) + kh;          // v2f column
      v2f ar = sAr[mw + ln][kk2];
      v2f ai = sAi[mw + ln][kk2];
      float2 w0 = sB[nw + ln][kk];
      float2 w1 = sB[nw + ln][kk + 1];
      v2f br  = {w0.x, w1.x};
      v2f bi  = {w0.y, w1.y};
      v2f nbi = -bi;
      accR = wmma_f32(ar, br,  accR);   // + xr*wr
      accR = wmma_f32(ai, nbi, accR);   // - xi*wi
      accI = wmma_f32(ar, bi,  accI);   // + xr*wi
      accI = wmma_f32(ai, br,  accI);   // + xi*wr
    }
    __syncthreads();
  }

  // ---- epilogue: conv bias + store (already the transposed final layout) ----
  const int e = n0 + nw + ln;
  const float2 bias = cb[e];
  const int mbase = m0 + mw + kh * 8;
#pragma unroll
  for (int r = 0; r < 8; ++r) {
    out[(size_t)(mbase + r) * pe::Ee + e] =
        make_float2(accR[r] + bias.x, accI[r] + bias.y);
  }
}

// ---------------------------------------------------------------------------
// K4: LN2 whitening + affine, in place over d_out [B, Hp*Wp, E] complex.
// 442368 elems per batch; 1728 blocks of 256 per batch (no batch straddle).
// ---------------------------------------------------------------------------
__global__ void pe_ln2_kernel(float2* __restrict__ out,
                              const float*  __restrict__ g2,  // [4][E,Hp,Wp]
                              const float2* __restrict__ b2,  // [E,Hp,Wp]
                              const float*  __restrict__ part2) {
  const int t   = threadIdx.x;
  const size_t gid = (size_t)blockIdx.x * 256 + t;
  const int b   = (int)(gid / pe::EHW);
  const int idx = (int)(gid % pe::EHW);
  const int patch = idx / pe::Ee;
  const int e     = idx % pe::Ee;

  __shared__ float sParams[5];
  if (t == 0) {
    float s[5] = {0.f, 0.f, 0.f, 0.f, 0.f};
    for (int i = 0; i < pe::NSPLIT; ++i)
#pragma unroll
      for (int q = 0; q < 5; ++q)
        s[q] += part2[(size_t)(b * pe::NSPLIT + i) * 5 + q];
    float mrr, mii, wrr, wii, wri;
    derive_whiten(s, 1.0f / (float)pe::EHW, mrr, mii, wrr, wii, wri);
    sParams[0] = mrr; sParams[1] = mii;
    sParams[2] = wrr; sParams[3] = wii; sParams[4] = wri;
  }
  __syncthreads();

  float2 y = out[gid];
  float xr = y.x - sParams[0];
  float xi = y.y - sParams[1];
  float zr = sParams[2] * xr + sParams[4] * xi;
  float zi = sParams[4] * xr + sParams[3] * xi;

  const size_t go = (size_t)e * pe::MPB + patch;   // params indexed [E,Hp,Wp]
  float g00 = g2[go];
  float g01 = g2[(size_t)pe::EHW + go];
  float g10 = g2[2 * (size_t)pe::EHW + go];
  float g11 = g2[3 * (size_t)pe::EHW + go];
  float2 bv = b2[go];
  out[gid] = make_float2(g00 * zr + g01 * zi + bv.x,
                         g10 * zr + g11 * zi + bv.y);
}

// ---------------------------------------------------------------------------
extern "C" void kernel_launch(void* const* d_in, const int* in_sizes, int n_in,
                              void* d_out, int out_size, void* d_ws, size_t ws_size,
                              hipStream_t stream) {
  const float2* x     = (const float2*)d_in[0];  // [B,C,H,W] complex64
  const float*  n1w   = (const float*) d_in[1];  // [2,2,C,H,W] f32
  const float2* n1b   = (const float2*)d_in[2];  // [C,H,W] complex64
  const float2* convw = (const float2*)d_in[3];  // [E,C,P,P] complex64 == [E,768]
  const float2* convb = (const float2*)d_in[4];  // [E] complex64
  const float*  n2w   = (const float*) d_in[5];  // [2,2,E,Hp,Wp] f32
  const float2* n2b   = (const float2*)d_in[6];  // [E,Hp,Wp] complex64
  float2* out = (float2*)d_out;                  // [B, Hp*Wp, E] complex64

  float* ws_f   = (float*)d_ws;
  float* part1  = ws_f;                                   // [64*16*5]
  float* part2  = ws_f + (size_t)pe::Bb * pe::NSPLIT * 5; // [64*16*5]

  // K1: LN1 moments over x
  pe_stats_kernel<<<pe::Bb * pe::NSPLIT, 256, 0, stream>>>(x, part1);

  // K2: fused LN1 + complex GEMM -> d_out (pre-LN2, final layout)
  const int grid_gemm = (pe::Mm / 32) * (pe::Ee / 64);    // 1152 * 12 = 13824
  pe_gemm_ln1_kernel<<<grid_gemm, 256, 0, stream>>>(x, n1w, n1b, convw, convb,
                                                    part1, out);

  // K3: LN2 moments over conv output (same slab size as K1)
  pe_stats_kernel<<<pe::Bb * pe::NSPLIT, 256, 0, stream>>>((const float2*)d_out,
                                                           part2);

  // K4: LN2 apply in place
  const int grid_ln2 = (pe::Bb * pe::EHW) / 256;          // 110592
  pe_ln2_kernel<<<grid_ln2, 256, 0, stream>>>(out, n2w, n2b, part2);
}